// DoublePAN_79199196938487
// MI455X (gfx1250) — compile-verified
//
#include <hip/hip_runtime.h>
#include <math.h>

// PANConv x2 for N=2048, E=32768, F_IN=512, H=16, C=32 on gfx1250.
// Strategy: never materialize dense A^2. pan^T @ u = w0*u + w1*A^T u + w2*A^T(A^T u).
// Bitset adjacency (dedupes duplicate edges, matching .set(1.0)); exact A^2 row
// counts via LDS scatter for deg; fp32 WMMA (v_wmma_f32_16x16x4_f32) for the
// dense linear layers.

typedef float v2f __attribute__((ext_vector_type(2)));
typedef float v8f __attribute__((ext_vector_type(8)));

#define N_NODES 2048
#define NWORDS  64      // 2048 / 32 bits
#define F_IN    512
#define H1      16
#define C2      32

// ---------------------------------------------------------------- bitmask build
__global__ void build_masks(const int* __restrict__ ei, unsigned* __restrict__ maskA,
                            unsigned* __restrict__ maskAT, int E) {
    int e = blockIdx.x * blockDim.x + threadIdx.x;
    if (e >= E) return;
    int s = ei[e];       // row  = source j
    int d = ei[E + e];   // col  = target i
    atomicOr(&maskA [s * NWORDS + (d >> 5)], 1u << (d & 31));
    atomicOr(&maskAT[d * NWORDS + (s >> 5)], 1u << (s & 31));
}

// ------------------------------------------------- exact deg (row-nnz of pan)
// One block per row r. cnt[c] = A^2[r,c] exactly (integer path counts), then
// deg = #{c : w0*I + w1*A + w2*A^2 != 0} for both filter-weight sets.
__global__ void deg_kernel(const unsigned* __restrict__ maskA,
                           const float* __restrict__ fw1, const float* __restrict__ fw2,
                           float* __restrict__ dinv1, float* __restrict__ dinv2) {
    __shared__ unsigned cnt[N_NODES];
    __shared__ unsigned rm[NWORDS];
    __shared__ unsigned nz1s, nz2s;
    int r = blockIdx.x;
    int t = threadIdx.x;                       // 256 threads
    for (int c = t; c < N_NODES; c += 256) cnt[c] = 0u;
    if (t < NWORDS) rm[t] = maskA[r * NWORDS + t];
    if (t == 0) { nz1s = 0u; nz2s = 0u; }
    __syncthreads();
    if (t < NWORDS) {                          // each thread owns one mask word
        unsigned word = rm[t];
        while (word) {                         // neighbor k of row r
            int k = t * 32 + __ffs(word) - 1;
            word &= word - 1;
            for (int w = 0; w < NWORDS; w++) {
                unsigned m2 = maskA[k * NWORDS + w];
                while (m2) {
                    int c = w * 32 + __ffs(m2) - 1;
                    m2 &= m2 - 1;
                    atomicAdd(&cnt[c], 1u);    // A^2[r,c] += 1
                }
            }
        }
    }
    __syncthreads();
    float w10 = fw1[0], w11 = fw1[1], w12 = fw1[2];
    float w20 = fw2[0], w21 = fw2[1], w22 = fw2[2];
    unsigned nz1 = 0u, nz2 = 0u;
    for (int c = t; c < N_NODES; c += 256) {
        float a  = ((rm[c >> 5] >> (c & 31)) & 1u) ? 1.0f : 0.0f;
        float a2 = (float)cnt[c];
        float id = (c == r) ? 1.0f : 0.0f;
        nz1 += ((w10 * id + w11 * a + w12 * a2) != 0.0f) ? 1u : 0u;
        nz2 += ((w20 * id + w21 * a + w22 * a2) != 0.0f) ? 1u : 0u;
    }
    atomicAdd(&nz1s, nz1);
    atomicAdd(&nz2s, nz2);
    __syncthreads();
    if (t == 0) {
        float d1 = (float)nz1s, d2 = (float)nz2s;
        dinv1[r] = (d1 > 0.0f) ? rsqrtf(fmaxf(d1, 1.0f)) : 0.0f;
        dinv2[r] = (d2 > 0.0f) ? rsqrtf(fmaxf(d2, 1.0f)) : 0.0f;
    }
}

// ------------------------------------------------------------- WMMA fp32 GEMM
// u[row, col] = dinv[row] * (xin[row,:K] . W[col,:K] + b[col])
// One wave per 16x16 tile; V_WMMA_F32_16X16X4_F32, K stepped by 4.
// 32-bit A layout: lanes 0-15 -> M=lane, K=0(v0)/1(v1); lanes 16-31 -> K=2/3.
// C/D layout: acc[e]: lanes 0-15 -> M=e, lanes 16-31 -> M=e+8; N=lane&15.
template <int K, int NOUT>
__global__ void gemm_wmma(const float* __restrict__ xin, const float* __restrict__ W,
                          const float* __restrict__ b, const float* __restrict__ dinv,
                          float* __restrict__ u) {
    int tileM = blockIdx.x;
    int tileN = blockIdx.y;
    int lane  = threadIdx.x;                   // 0..31, one full wave32
    int row   = tileM * 16 + (lane & 15);
    int col   = tileN * 16 + (lane & 15);
    int khalf = (lane >> 4) << 1;              // 0 for lanes 0-15, 2 for 16-31
    const float* xr = xin + row * K;
    const float* wr = W   + col * K;           // W is [NOUT, K] row-major -> W^T[k,col]
    v8f acc = {};
    for (int k = 0; k < K; k += 4) {
        v2f a, bb;
        a.x  = xr[k + khalf];     a.y  = xr[k + khalf + 1];
        bb.x = wr[k + khalf];     bb.y = wr[k + khalf + 1];
        acc = __builtin_amdgcn_wmma_f32_16x16x4_f32(
            /*neg_a=*/false, a, /*neg_b=*/false, bb,
            /*c_mod=*/(short)0, acc, /*reuse_a=*/false, /*reuse_b=*/false);
    }
    int rbase  = tileM * 16 + ((lane >> 4) << 3);
    float bias = b[col];
    #pragma unroll
    for (int e = 0; e < 8; e++) {
        int rr = rbase + e;
        u[rr * NOUT + col] = dinv[rr] * (acc[e] + bias);
    }
}

// ------------------------------------------------- A^T sparse matvec (bitset)
// out[i, c] = sum_{j : A[j,i]=1} u[j, c], via row bitmasks of A^T.
template <int F>
__global__ void at_spmv(const unsigned* __restrict__ maskAT,
                        const float* __restrict__ u, float* __restrict__ out) {
    const int R = 256 / F;                     // rows per block
    __shared__ unsigned sm[R][NWORDS];
    int c  = threadIdx.x % F;
    int rl = threadIdx.x / F;
    int i  = blockIdx.x * R + rl;
    for (int w = c; w < NWORDS; w += F) sm[rl][w] = maskAT[i * NWORDS + w];
    __syncthreads();
    float acc = 0.0f;
    for (int w = 0; w < NWORDS; w++) {
        unsigned word = sm[rl][w];
        while (word) {
            int j = w * 32 + __ffs(word) - 1;
            word &= word - 1;
            acc += u[j * F + c];
        }
    }
    out[i * F + c] = acc;
}

// ---------------------------------------------------------------- combine (+relu)
template <int F, bool RELU>
__global__ void combine(const float* __restrict__ u,  const float* __restrict__ s1,
                        const float* __restrict__ s2, const float* __restrict__ fw,
                        const float* __restrict__ dinv, float* __restrict__ out) {
    int idx = blockIdx.x * blockDim.x + threadIdx.x;
    if (idx >= N_NODES * F) return;
    int i = idx / F;
    float v = dinv[i] * (fw[0] * u[idx] + fw[1] * s1[idx] + fw[2] * s2[idx]);
    if (RELU) v = fmaxf(v, 0.0f);
    out[idx] = v;
}

// --------------------------------------------------------- log_softmax (32 cols)
__global__ void log_softmax32(const float* __restrict__ in, float* __restrict__ out) {
    int r = blockIdx.x;
    int c = threadIdx.x;                       // wave32 == 32 classes
    float v = in[r * 32 + c];
    float m = v;
    #pragma unroll
    for (int o = 16; o > 0; o >>= 1) m = fmaxf(m, __shfl_xor(m, o, 32));
    float e = expf(v - m);
    float s = e;
    #pragma unroll
    for (int o = 16; o > 0; o >>= 1) s += __shfl_xor(s, o, 32);
    out[r * 32 + c] = (v - m) - logf(s);
}

// ----------------------------------------------------------------------- launch
extern "C" void kernel_launch(void* const* d_in, const int* in_sizes, int n_in,
                              void* d_out, int out_size, void* d_ws, size_t ws_size,
                              hipStream_t stream) {
    const float* x      = (const float*)d_in[0];
    const int*   ei     = (const int*)  d_in[1];
    const float* fw1    = (const float*)d_in[2];
    const float* lin1_w = (const float*)d_in[3];
    const float* lin1_b = (const float*)d_in[4];
    const float* fw2    = (const float*)d_in[5];
    const float* lin2_w = (const float*)d_in[6];
    const float* lin2_b = (const float*)d_in[7];
    int E = in_sizes[1] / 2;

    char* p = (char*)d_ws;
    unsigned* maskA  = (unsigned*)p; p += (size_t)N_NODES * NWORDS * 4;
    unsigned* maskAT = (unsigned*)p; p += (size_t)N_NODES * NWORDS * 4;
    float* dinv1 = (float*)p; p += N_NODES * 4;
    float* dinv2 = (float*)p; p += N_NODES * 4;
    float* u1  = (float*)p; p += N_NODES * H1 * 4;
    float* s1a = (float*)p; p += N_NODES * H1 * 4;
    float* s1b = (float*)p; p += N_NODES * H1 * 4;
    float* r1  = (float*)p; p += N_NODES * H1 * 4;
    float* u2  = (float*)p; p += N_NODES * C2 * 4;
    float* s2a = (float*)p; p += N_NODES * C2 * 4;
    float* s2b = (float*)p; p += N_NODES * C2 * 4;
    float* c2  = (float*)p; p += N_NODES * C2 * 4;

    // maskA and maskAT are contiguous -> single memset (graph-capturable)
    hipMemsetAsync(maskA, 0, (size_t)2 * N_NODES * NWORDS * 4, stream);
    build_masks<<<(E + 255) / 256, 256, 0, stream>>>(ei, maskA, maskAT, E);
    deg_kernel<<<N_NODES, 256, 0, stream>>>(maskA, fw1, fw2, dinv1, dinv2);

    // Layer 1: u1 = dinv1 * (x @ W1^T + b1); s = A^T u; out = relu(dinv1 * pan^T u)
    gemm_wmma<F_IN, H1><<<dim3(N_NODES / 16, 1), 32, 0, stream>>>(x, lin1_w, lin1_b, dinv1, u1);
    at_spmv<H1><<<N_NODES / (256 / H1), 256, 0, stream>>>(maskAT, u1, s1a);
    at_spmv<H1><<<N_NODES / (256 / H1), 256, 0, stream>>>(maskAT, s1a, s1b);
    combine<H1, true><<<(N_NODES * H1 + 255) / 256, 256, 0, stream>>>(u1, s1a, s1b, fw1, dinv1, r1);

    // Layer 2
    gemm_wmma<H1, C2><<<dim3(N_NODES / 16, C2 / 16), 32, 0, stream>>>(r1, lin2_w, lin2_b, dinv2, u2);
    at_spmv<C2><<<N_NODES / (256 / C2), 256, 0, stream>>>(maskAT, u2, s2a);
    at_spmv<C2><<<N_NODES / (256 / C2), 256, 0, stream>>>(maskAT, s2a, s2b);
    combine<C2, false><<<(N_NODES * C2 + 255) / 256, 256, 0, stream>>>(u2, s2a, s2b, fw2, dinv2, c2);

    log_softmax32<<<N_NODES, 32, 0, stream>>>(c2, (float*)d_out);
}